// RandomCrop_10677288698084
// MI455X (gfx1250) — compile-verified
//
#include <hip/hip_runtime.h>

// RandomCrop / integer shift with zero padding.
// out[n,c,i,j] = in[n,c,i+dy,j+dx] (zeros out of range), dx,dy in [-8,8].
// Pure data movement: 192 MB total traffic -> ~8.3 us floor at 23.3 TB/s.
// Uses CDNA5 async global<->LDS DMA path (ASYNCcnt) instead of VGPR staging.

#define MAXP 8

typedef __attribute__((address_space(1))) int gint_t;   // global b32 word
typedef __attribute__((address_space(3))) int lint_t;   // LDS b32 word
typedef __attribute__((address_space(3))) void lvoid_t;

#if defined(__has_builtin)
#  if __has_builtin(__builtin_amdgcn_global_load_async_to_lds_b32) && \
      __has_builtin(__builtin_amdgcn_global_store_async_from_lds_b32)
#    define USE_ASYNC_BUILTINS 1
#  endif
#  if __has_builtin(__builtin_amdgcn_s_wait_asynccnt)
#    define USE_WAIT_BUILTIN 1
#  endif
#endif

__device__ __forceinline__ void async_wait0() {
#if defined(USE_WAIT_BUILTIN)
  __builtin_amdgcn_s_wait_asynccnt(0);
#else
  asm volatile("s_wait_asynccnt 0" ::: "memory");
#endif
}

__device__ __forceinline__ void async_g2l_b32(const float* g, float* l) {
#if defined(USE_ASYNC_BUILTINS)
  __builtin_amdgcn_global_load_async_to_lds_b32((gint_t*)g, (lint_t*)l, 0, 0);
#else
  unsigned loff = (unsigned)(unsigned long long)(lvoid_t*)l;  // LDS byte offset
  unsigned long long ga = (unsigned long long)g;
  asm volatile("global_load_async_to_lds_b32 %0, %1, off"
               :: "v"(loff), "v"(ga) : "memory");
#endif
}

__device__ __forceinline__ void async_l2g_b32(float* g, const float* l) {
#if defined(USE_ASYNC_BUILTINS)
  __builtin_amdgcn_global_store_async_from_lds_b32((gint_t*)g, (lint_t*)l, 0, 0);
#else
  unsigned loff = (unsigned)(unsigned long long)(lvoid_t*)l;
  unsigned long long ga = (unsigned long long)g;
  asm volatile("global_store_async_from_lds_b32 %1, %0, off"
               :: "v"(loff), "v"(ga) : "memory");
#endif
}

// One block = one output row (n,c,i); 256 threads = 8 wave32.
__global__ __launch_bounds__(256)
void shift_rows_async(const float* __restrict__ in,
                      const int* __restrict__ shifts,
                      float* __restrict__ out) {
  constexpr int W = 256;
  constexpr int H = 256;

  const int row = blockIdx.x;         // 0 .. N*C*H-1
  const int j   = threadIdx.x;        // 0 .. 255 (column)
  const int i   = row & (H - 1);      // row within image plane
  const int nc  = row >> 8;           // n*C + c  (uniform per block)
  const int n   = nc / 3;             // uniform -> scalar path

  // shifts[n] = {raw_dx, raw_dy}; reference subtracts MAXP.
  const int dx = shifts[2 * n + 0] - MAXP;
  const int dy = shifts[2 * n + 1] - MAXP;

  float* orow = out + (size_t)row * W;

  const int si = i + dy;
  if ((unsigned)si >= (unsigned)H) {   // whole source row out of range -> zeros
    orow[j] = 0.0f;
    return;
  }

  const float* irow = in + ((size_t)nc * H + si) * W;

  // valid output columns: j in [j0, j1)
  const int j0 = (dx < 0) ? -dx : 0;
  const int j1 = (dx > 0) ? (W - dx) : W;

  __shared__ float lbuf[W];

  if (j >= j0 && j < j1) {
    // memory -> LDS -> memory, per-lane b32 (4B-aligned regardless of dx),
    // each lane owns exactly LDS word j: no cross-wave dependency, no barrier.
    async_g2l_b32(irow + (j + dx), &lbuf[j]);
    async_wait0();                     // load landed in LDS
    async_l2g_b32(orow + j, &lbuf[j]);
    // completion guaranteed by implicit wait-idle at s_endpgm
  } else {
    orow[j] = 0.0f;                    // horizontal zero padding (<= 8 lanes)
  }
}

extern "C" void kernel_launch(void* const* d_in, const int* in_sizes, int n_in,
                              void* d_out, int out_size, void* d_ws, size_t ws_size,
                              hipStream_t stream) {
  const float* in     = (const float*)d_in[0];   // [N,C,H,W] fp32
  const int*   shifts = (const int*)d_in[1];     // [N,2] int32
  float*       out    = (float*)d_out;

  const int rows = in_sizes[0] / 256;            // N*C*H (W = 256)
  shift_rows_async<<<rows, 256, 0, stream>>>(in, shifts, out);
}